// AbstractContextLayer_63350767616448
// MI455X (gfx1250) — compile-verified
//
#include <hip/hip_runtime.h>

typedef __attribute__((ext_vector_type(2))) float v2f;
typedef __attribute__((ext_vector_type(4))) float f4;
typedef __attribute__((ext_vector_type(8))) float v8f;
typedef __attribute__((ext_vector_type(4))) int   i4;

// One wave32 processes a tile of 16 entries:
//   lane L: entry m = L&15, neighbor-half = L>>4 (neighbors 0-3 or 4-7)
//   gather 4 neighbor vectors (8 f32 = 2x b128 each), accumulate,
//   butterfly-add across half-waves, then 2x v_wmma_f32_16x16x4_f32
//   computes D[16x16] = Y[16x8] * (W^T/8)[8x16(pad)] + bias.
__global__ __launch_bounds__(256)
void hashgrid_avg_linear_wmma(const float* __restrict__ x,
                              const float* __restrict__ W,
                              const float* __restrict__ bias,
                              const int*   __restrict__ nidx,
                              float*       __restrict__ out,
                              int N)
{
    const int lane = threadIdx.x & 31;
    const int wave = threadIdx.x >> 5;
    const int tile = blockIdx.x * 8 + wave;     // 8 waves per block
    const int n0   = tile * 16;                  // first entry of tile (wave-uniform)
    if (n0 >= N) return;                         // uniform per wave -> EXEC stays full

    const int m    = lane & 15;                  // entry within tile
    const int half = lane >> 4;                  // 0: neighbors 0-3, 1: neighbors 4-7

    // ---- neighbor indices: stream-once, non-temporal, fully coalesced ----
    int e = n0 + m;
    if (e >= N) e = N - 1;                       // clamp tail (stores masked below)
    const i4 nb = __builtin_nontemporal_load(
        reinterpret_cast<const i4*>(nidx + (size_t)e * 8 + (size_t)half * 4));

    // ---- gather 4 neighbor feature vectors and accumulate ----
    f4 accA = (f4)0.0f, accB = (f4)0.0f;
    {
        const f4* p0 = reinterpret_cast<const f4*>(x + (size_t)nb.x * 8);
        const f4* p1 = reinterpret_cast<const f4*>(x + (size_t)nb.y * 8);
        const f4* p2 = reinterpret_cast<const f4*>(x + (size_t)nb.z * 8);
        const f4* p3 = reinterpret_cast<const f4*>(x + (size_t)nb.w * 8);
        accA += p0[0]; accB += p0[1];
        accA += p1[0]; accB += p1[1];
        accA += p2[0]; accB += p2[1];
        accA += p3[0]; accB += p3[1];
    }

    // ---- merge half-wave partial sums (neighbors 0-3 + 4-7) ----
    float acc[8] = { accA.x, accA.y, accA.z, accA.w,
                     accB.x, accB.y, accB.z, accB.w };
#pragma unroll
    for (int c = 0; c < 8; ++c)
        acc[c] += __shfl_xor(acc[c], 16, 32);
    // every lane now holds the full 8-channel neighbor sum for entry m

    // ---- B matrix: W^T * (1/8), zero-padded N=8..15; C preloaded with bias ----
    // B 4x16 f32 layout (mirrors A): VGPR0: lanes0-15 K=0 / lanes16-31 K=2;
    //                                 VGPR1: lanes0-15 K=1 / lanes16-31 K=3.
    const int ocol = lane & 15;                  // output channel column
    float b0x = 0.f, b0y = 0.f, b1x = 0.f, b1y = 0.f, c_init = 0.f;
    if (ocol < 8) {
        const float s  = 0.125f;                 // fold the 1/8 average into W
        const int   kb = half * 2;               // K base for this half-wave
        const float* wr = W + ocol * 8;          // row o of W (contiguous in c)
        b0x = wr[kb + 0] * s;                    // WMMA #1: channels 0-3
        b0y = wr[kb + 1] * s;
        b1x = wr[4 + kb + 0] * s;                // WMMA #2: channels 4-7
        b1y = wr[4 + kb + 1] * s;
        c_init = bias[ocol];
    }
    v2f B0; B0.x = b0x; B0.y = b0y;
    v2f B1; B1.x = b1x; B1.y = b1y;

    // ---- A operands: lanes 0-15 carry K={0,1}/{4,5}; lanes 16-31 K={2,3}/{6,7} ----
    v2f A0, A1;
    A0.x = half ? acc[2] : acc[0];
    A0.y = half ? acc[3] : acc[1];
    A1.x = half ? acc[6] : acc[4];
    A1.y = half ? acc[7] : acc[5];

    v8f C;
#pragma unroll
    for (int r = 0; r < 8; ++r) C[r] = c_init;   // bias broadcast over M rows

    // D = A1*B1 + (A0*B0 + bias): two chained K=4 steps cover C=8
    C = __builtin_amdgcn_wmma_f32_16x16x4_f32(false, A0, false, B0,
                                              (short)0, C, false, false);
    C = __builtin_amdgcn_wmma_f32_16x16x4_f32(false, A1, false, B1,
                                              (short)0, C, false, false);

    // ---- store: VGPR r holds M = r + 8*half, N = ocol ----
    // Wave-uniform full-tile fast path: one base pointer per lane, 8 stores
    // with immediate offsets (r*32B). Only the single tail tile predicates.
    if (ocol < 8) {
        float* p = out + (size_t)(n0 + half * 8) * 8 + ocol;
        if (n0 + 16 <= N) {                      // full tile (wave-uniform)
#pragma unroll
            for (int r = 0; r < 8; ++r)
                __builtin_nontemporal_store(C[r], p + r * 8);
        } else {                                 // tail tile only
#pragma unroll
            for (int r = 0; r < 8; ++r)
                if (n0 + half * 8 + r < N)
                    __builtin_nontemporal_store(C[r], p + r * 8);
        }
    }
}

extern "C" void kernel_launch(void* const* d_in, const int* in_sizes, int n_in,
                              void* d_out, int out_size, void* d_ws, size_t ws_size,
                              hipStream_t stream)
{
    const float* x    = (const float*)d_in[0];   // (N, 1, 8) f32
    const float* W    = (const float*)d_in[1];   // (8, 8)    f32
    const float* bias = (const float*)d_in[2];   // (8,)      f32
    const int*   nidx = (const int*)d_in[3];     // (N, 8)    i32

    float* out = (float*)d_out;                  // (N, 1, 8) f32
    const int N      = in_sizes[3] / 8;
    const int tiles  = (N + 15) / 16;            // 16 entries per wave
    const int blocks = (tiles + 7) / 8;          // 8 waves per 256-thread block

    hipLaunchKernelGGL(hashgrid_avg_linear_wmma,
                       dim3(blocks), dim3(256), 0, stream,
                       x, W, bias, nidx, out, N);
}